// RAConv_14645838479438
// MI455X (gfx1250) — compile-verified
//
#include <hip/hip_runtime.h>

#define NNODES 100000
#define NEDGES 800000
#define CCH    128            // channels
#define KTOT   384            // 3 * CCH concatenated K dimension

typedef __attribute__((ext_vector_type(2))) float v2f;
typedef __attribute__((ext_vector_type(8))) float v8f;

// ---------- monotonic float<->uint encoding for atomicMax-based segment max ----------
__device__ __forceinline__ unsigned enc_f32(float f) {
    unsigned u = __float_as_uint(f);
    return (u & 0x80000000u) ? ~u : (u | 0x80000000u);
}
__device__ __forceinline__ float dec_f32(unsigned u) {
    return __uint_as_float((u & 0x80000000u) ? (u ^ 0x80000000u) : ~u);
}

// ---------- workspace zeroing ----------
__global__ __launch_bounds__(256) void zero_kernel(unsigned* __restrict__ p, size_t n) {
    size_t i = (size_t)blockIdx.x * blockDim.x + threadIdx.x;
    size_t stride = (size_t)gridDim.x * blockDim.x;
    for (; i < n; i += stride) p[i] = 0u;
}

// ---------- pass 1: per-edge logits + segment max (one wave32 per edge) ----------
__global__ __launch_bounds__(256) void edge_logits_kernel(
    const float* __restrict__ x, const int* __restrict__ src, const int* __restrict__ dst,
    float* __restrict__ logits, unsigned* __restrict__ segmax)
{
    const int e = blockIdx.x * 8 + (threadIdx.x >> 5);
    if (e >= NEDGES) return;
    const int lane = threadIdx.x & 31;
    const int s = src[e], d = dst[e];
    const float4 a = ((const float4*)(x + (size_t)s * CCH))[lane];
    const float4 b = ((const float4*)(x + (size_t)d * CCH))[lane];
    float p = a.x * b.x + a.y * b.y + a.z * b.z + a.w * b.w;
    #pragma unroll
    for (int off = 16; off > 0; off >>= 1) p += __shfl_xor(p, off, 32);
    if (lane == 0) {
        const float logit = p * 0.08838834764831845f;   // 1/sqrt(128)
        logits[e] = logit;
        atomicMax(segmax + d, enc_f32(logit));
    }
}

// ---------- pass 2: e = exp(logit - max); denom, cnt, sum_x scatters ----------
__global__ __launch_bounds__(256) void edge_expsum_kernel(
    const float* __restrict__ x, const int* __restrict__ src, const int* __restrict__ dst,
    float* __restrict__ ebuf /* in: logit, out: exp */, const unsigned* __restrict__ segmax,
    float* __restrict__ denom, float* __restrict__ cnt, float* __restrict__ sumx)
{
    const int e = blockIdx.x * 8 + (threadIdx.x >> 5);
    if (e >= NEDGES) return;
    const int lane = threadIdx.x & 31;
    const int s = src[e], d = dst[e];
    const float m = dec_f32(segmax[d]);
    const float ev = __expf(ebuf[e] - m);
    if (lane == 0) {
        ebuf[e] = ev;
        atomicAdd(denom + d, ev);
        atomicAdd(cnt + d, 1.0f);
    }
    const float4 a = ((const float4*)(x + (size_t)s * CCH))[lane];
    float* sx = sumx + (size_t)d * CCH + lane * 4;
    atomicAdd(sx + 0, a.x); atomicAdd(sx + 1, a.y);
    atomicAdd(sx + 2, a.z); atomicAdd(sx + 3, a.w);
}

// ---------- pass 3: msg += attn * x_src ; var_acc += (x_src - mean)^2 ----------
__global__ __launch_bounds__(256) void edge_msgvar_kernel(
    const float* __restrict__ x, const int* __restrict__ src, const int* __restrict__ dst,
    const float* __restrict__ ebuf, const float* __restrict__ denom,
    const float* __restrict__ cnt, const float* __restrict__ sumx,
    float* __restrict__ msg, float* __restrict__ var)
{
    const int e = blockIdx.x * 8 + (threadIdx.x >> 5);
    if (e >= NEDGES) return;
    const int lane = threadIdx.x & 31;
    const int s = src[e], d = dst[e];
    const float attn = ebuf[e] / denom[d];
    const float rc = 1.0f / fmaxf(cnt[d], 1.0f);
    const float4 a  = ((const float4*)(x    + (size_t)s * CCH))[lane];
    const float4 sm = ((const float4*)(sumx + (size_t)d * CCH))[lane];
    const float dx0 = a.x - sm.x * rc, dx1 = a.y - sm.y * rc;
    const float dx2 = a.z - sm.z * rc, dx3 = a.w - sm.w * rc;
    float* mp = msg + (size_t)d * CCH + lane * 4;
    float* vp = var + (size_t)d * CCH + lane * 4;
    atomicAdd(mp + 0, attn * a.x); atomicAdd(mp + 1, attn * a.y);
    atomicAdd(mp + 2, attn * a.z); atomicAdd(mp + 3, attn * a.w);
    atomicAdd(vp + 0, dx0 * dx0);  atomicAdd(vp + 1, dx1 * dx1);
    atomicAdd(vp + 2, dx2 * dx2);  atomicAdd(vp + 3, dx3 * dx3);
}

// ---------- fused WMMA GEMM: out = [x|msg|var/cnt] @ [Ws;Wn;Wv] + (bs+bn+bv) ----------
// One wave32 computes a 16x16 output tile with V_WMMA_F32_16X16X4_F32 (full f32 path).
// A frag (16x4): lane<16 holds A[m=lane][k..k+1], lane>=16 holds A[m=lane-16][k+2..k+3].
// B frag (4x16): lane<16 holds B[k..k+1][n=lane], lane>=16 holds B[k+2..k+3][n=lane-16].
// C/D: VGPR j -> M = j + 8*(lane>=16), N = lane&15.
__device__ __forceinline__ v8f gemm_chunk(v8f acc, const float* __restrict__ Arow, float scl,
                                          const float* __restrict__ Bpanel, int lr, int hl)
{
    #pragma unroll 8
    for (int k = 0; k < CCH; k += 4) {
        const int kk = k + 2 * hl;
        v2f a = *(const v2f*)(Arow + kk);
        a *= scl;
        v2f b;
        b.x = Bpanel[kk * 16 + lr];
        b.y = Bpanel[(kk + 1) * 16 + lr];
        acc = __builtin_amdgcn_wmma_f32_16x16x4_f32(false, a, false, b, (short)0, acc,
                                                    false, false);
    }
    return acc;
}

__global__ __launch_bounds__(256) void raconv_gemm_kernel(
    const float* __restrict__ x, const float* __restrict__ msg, const float* __restrict__ var,
    const float* __restrict__ cnt,
    const float* __restrict__ w_self, const float* __restrict__ w_neigh, const float* __restrict__ w_var,
    const float* __restrict__ b_self, const float* __restrict__ b_neigh, const float* __restrict__ b_var,
    float* __restrict__ out)
{
    __shared__ float ldsB[KTOT * 16];   // 24 KB B panel: [k 0..383][j 0..15]
    const int n0 = blockIdx.y * 16;
    const int m_block = blockIdx.x * 128;
    const int tid = threadIdx.x;

    // stage W[:, n0:n0+16] for all three weight matrices into LDS (shared by 8 waves)
    for (int i = tid; i < KTOT * 16; i += 256) {
        const int k = i >> 4, j = i & 15;
        const float* W = (k < CCH) ? w_self : ((k < 2 * CCH) ? w_neigh : w_var);
        ldsB[i] = W[(k & (CCH - 1)) * CCH + n0 + j];
    }
    __syncthreads();

    const int wave = tid >> 5;
    const int lane = tid & 31;
    const int hl = lane >> 4;       // half-wave selector (K split)
    const int lr = lane & 15;       // row/col within tile
    const int m0 = m_block + wave * 16;
    int mrow = m0 + lr;
    if (mrow >= NNODES) mrow = NNODES - 1;   // clamp (keeps EXEC all-ones for WMMA)
    const float rc = 1.0f / fmaxf(cnt[mrow], 1.0f);

    v8f acc = {};
    acc = gemm_chunk(acc, x   + (size_t)mrow * CCH, 1.0f, ldsB,               lr, hl);
    acc = gemm_chunk(acc, msg + (size_t)mrow * CCH, 1.0f, ldsB + 1 * CCH * 16, lr, hl);
    acc = gemm_chunk(acc, var + (size_t)mrow * CCH, rc,   ldsB + 2 * CCH * 16, lr, hl);

    const float bias = b_self[n0 + lr] + b_neigh[n0 + lr] + b_var[n0 + lr];
    #pragma unroll
    for (int j = 0; j < 8; ++j) {
        const int row = m0 + j + 8 * hl;
        if (row < NNODES) out[(size_t)row * CCH + n0 + lr] = acc[j] + bias;
    }
}

extern "C" void kernel_launch(void* const* d_in, const int* in_sizes, int n_in,
                              void* d_out, int out_size, void* d_ws, size_t ws_size,
                              hipStream_t stream) {
    const float* x       = (const float*)d_in[0];
    const float* w_self  = (const float*)d_in[1];
    const float* b_self  = (const float*)d_in[2];
    const float* w_neigh = (const float*)d_in[3];
    const float* b_neigh = (const float*)d_in[4];
    const float* w_var   = (const float*)d_in[5];
    const float* b_var   = (const float*)d_in[6];
    const int*   eidx    = (const int*)d_in[7];
    const int* src = eidx;              // edge_index[0]
    const int* dst = eidx + NEDGES;     // edge_index[1]
    float* out = (float*)d_out;

    // workspace layout (floats): logits[E] | segmax[N] | denom[N] | cnt[N] |
    //                            sumx[N*C] | msg[N*C] | var[N*C]   (~158 MB)
    float* ws       = (float*)d_ws;
    float* logits   = ws;
    unsigned* segmax = (unsigned*)(ws + NEDGES);
    float* denom    = ws + NEDGES + NNODES;
    float* cnt      = denom + NNODES;
    float* sumx     = cnt + NNODES;
    float* msg      = sumx + (size_t)NNODES * CCH;
    float* var      = msg  + (size_t)NNODES * CCH;

    const size_t zero_words = (size_t)3 * NNODES + (size_t)3 * NNODES * CCH;
    zero_kernel<<<2048, 256, 0, stream>>>((unsigned*)segmax, zero_words);

    edge_logits_kernel<<<NEDGES / 8, 256, 0, stream>>>(x, src, dst, logits, segmax);
    edge_expsum_kernel<<<NEDGES / 8, 256, 0, stream>>>(x, src, dst, logits, segmax,
                                                       denom, cnt, sumx);
    edge_msgvar_kernel<<<NEDGES / 8, 256, 0, stream>>>(x, src, dst, logits, denom,
                                                       cnt, sumx, msg, var);

    dim3 grid((NNODES + 127) / 128, CCH / 16);
    raconv_gemm_kernel<<<grid, 256, 0, stream>>>(x, msg, var, cnt,
                                                 w_self, w_neigh, w_var,
                                                 b_self, b_neigh, b_var, out);
}